// FaceBoxesDetect_16243566313648
// MI455X (gfx1250) — compile-verified
//
#include <hip/hip_runtime.h>
#include <hip/hip_bf16.h>
#include <stdint.h>

// ---------------- problem constants ----------------
#define BATCH       16
#define NPRIOR      21824
#define NCAND       1024          // KCAND
#define NTOPK       750
#define CONF_TH     0.3f
#define IOU_TH      0.01f
#define VAR0        0.1f
#define VAR1        0.2f
#define IMG_SCALE   1024.0f

typedef float    v2f  __attribute__((ext_vector_type(2)));
typedef float    v8f  __attribute__((ext_vector_type(8)));
typedef uint64_t u64;
typedef uint32_t u32;
typedef uint16_t u16;

// Monotone map: float -> u32 such that ascending u32 == DESCENDING float.
__device__ __forceinline__ u32 flip_desc(float f) {
    u32 u = __float_as_uint(f);
    return (u & 0x80000000u) ? u : (~u & 0x7FFFFFFFu);
}
// key: (score desc, idx asc) ascending-u64 order; unique because idx unique.
__device__ __forceinline__ u64 make_key(float s, u32 idx) {
    return ((u64)flip_desc(s) << 32) | (u64)idx;
}

// ---------------- kernel 1: decode boxes + sigmoid score ----------------
__global__ void k_decode_score(const float* __restrict__ boxes_logits,
                               const float* __restrict__ cls_logits,
                               const float* __restrict__ priors,
                               float* __restrict__ boxesDec,    // [B,P,4]
                               float* __restrict__ scoresM) {   // [B,P] masked
    int i = blockIdx.x * blockDim.x + threadIdx.x;
    if (i >= BATCH * NPRIOR) return;
    int p = i % NPRIOR;

    float lx = boxes_logits[i * 4 + 0];
    float ly = boxes_logits[i * 4 + 1];
    float lw = boxes_logits[i * 4 + 2];
    float lh = boxes_logits[i * 4 + 3];
    float px = priors[p * 4 + 0];
    float py = priors[p * 4 + 1];
    float pw = priors[p * 4 + 2];
    float ph = priors[p * 4 + 3];

    float cx = px + lx * VAR0 * pw;
    float cy = py + ly * VAR0 * ph;
    float w  = pw * __expf(lw * VAR1);
    float h  = ph * __expf(lh * VAR1);
    float x1 = cx - 0.5f * w;
    float y1 = cy - 0.5f * h;
    float x2 = x1 + w;
    float y2 = y1 + h;
    boxesDec[i * 4 + 0] = x1 * IMG_SCALE;
    boxesDec[i * 4 + 1] = y1 * IMG_SCALE;
    boxesDec[i * 4 + 2] = x2 * IMG_SCALE;
    boxesDec[i * 4 + 3] = y2 * IMG_SCALE;

    // softmax over 2 classes, class-1 prob == sigmoid(l1 - l0)
    float l0 = cls_logits[i * 2 + 0];
    float l1 = cls_logits[i * 2 + 1];
    float s  = 1.0f / (1.0f + __expf(l0 - l1));
    scoresM[i] = (s > CONF_TH) ? s : -1.0f;   // reference masks before top_k
}

// ---------------- kernel 2: per-batch exact top-1024 (radix-select + bitonic) ----
__global__ __launch_bounds__(256) void k_select(const float* __restrict__ scoresM,
                                                const float* __restrict__ boxesDec,
                                                float* __restrict__ candBoxes,   // [B,1024,4]
                                                float* __restrict__ candScores,  // [B,1024]
                                                float* __restrict__ candArea) {  // [B,1024]
    __shared__ u32 bins[256];
    __shared__ u64 selT;
    __shared__ u32 rem_s;
    __shared__ u32 cnt;
    __shared__ u64 sel[NCAND];

    const int b   = blockIdx.x;
    const int tid = threadIdx.x;
    const float* sc = scoresM + (size_t)b * NPRIOR;

    if (tid == 0) { selT = 0; rem_s = NCAND; cnt = 0; }
    __syncthreads();

    // 8 byte-passes MSB->LSB: find exact 1024-th smallest 64-bit key.
    for (int pass = 7; pass >= 0; --pass) {
        bins[tid] = 0;
        __syncthreads();
        const int sh  = pass * 8;
        const u64 hiC = selT;
        for (int i = tid; i < NPRIOR; i += 256) {
            u64 k = make_key(sc[i], (u32)i);
            bool ok = (pass == 7) || ((k >> (sh + 8)) == (hiC >> (sh + 8)));
            if (ok) atomicAdd(&bins[(u32)(k >> sh) & 255u], 1u);
        }
        __syncthreads();
        if (tid == 0) {
            u32 remc = rem_s, cum = 0;
            for (u32 v = 0; v < 256; ++v) {
                u32 c = bins[v];
                if (cum + c >= remc) {
                    selT  = hiC | ((u64)v << sh);
                    rem_s = remc - cum;
                    break;
                }
                cum += c;
            }
        }
        __syncthreads();
    }
    const u64 T = selT;   // exact 1024-th smallest key (keys unique)

    // compact: keys <= T are exactly the 1024 candidates (order arbitrary here)
    for (int i = tid; i < NPRIOR; i += 256) {
        u64 k = make_key(sc[i], (u32)i);
        if (k <= T) {
            u32 p = atomicAdd(&cnt, 1u);
            if (p < NCAND) sel[p] = k;
        }
    }
    __syncthreads();

    // bitonic sort 1024 keys ascending  ==  score desc, idx asc (jax top_k order)
    for (u32 k2 = 2; k2 <= NCAND; k2 <<= 1) {
        for (u32 j = k2 >> 1; j > 0; j >>= 1) {
            for (u32 t = tid; t < NCAND / 2; t += 256) {
                u32 i  = ((t & ~(j - 1)) << 1) | (t & (j - 1));
                u32 ip = i | j;
                bool up = ((i & k2) == 0);
                u64 a = sel[i], c = sel[ip];
                if ((a > c) == up) { sel[i] = c; sel[ip] = a; }
            }
            __syncthreads();
        }
    }

    // gather candidate boxes / scores / areas
    for (int t = tid; t < NCAND; t += 256) {
        u64 k   = sel[t];
        u32 idx = (u32)k;
        float s = sc[idx];
        candScores[b * NCAND + t] = s;
        const float* bx = boxesDec + ((size_t)b * NPRIOR + idx) * 4;
        float x1 = bx[0], y1 = bx[1], x2 = bx[2], y2 = bx[3];
        float* cb = candBoxes + ((size_t)b * NCAND + t) * 4;
        cb[0] = x1; cb[1] = y1; cb[2] = x2; cb[3] = y2;
        candArea[b * NCAND + t] = (x2 - x1) * (y2 - y1);   // unclamped, as reference
    }
}

// ---------------- kernel 3: IoU>thresh bit-matrix, one wave per 16x16 tile ------
// union = (area_i + area_j) - inter.  The pairwise sum matrix is rank-2:
//   [area_m, 1] (16x4, K padded) x [[1..1],[area_n..]] (4x16)  -> one WMMA tile.
__global__ __launch_bounds__(32) void k_iou_tiles(const float* __restrict__ candBoxes,
                                                  const float* __restrict__ candArea,
                                                  u16* __restrict__ bitmat16) { // [B,1024,64]
    const int tm = blockIdx.x;      // row tile   [0,64)
    const int tn = blockIdx.y;      // col tile   [0,64)
    const int b  = blockIdx.z;      // batch
    const int lane = threadIdx.x;   // wave32

    const float* area = candArea + (size_t)b * NCAND;

    // A (16x4, MxK): lanes 0-15 hold M=lane,K={0,1}; lanes 16-31 hold M=lane-16,K={2,3}
    v2f a;
    if (lane < 16) { a.x = area[tm * 16 + lane]; a.y = 1.0f; }
    else           { a.x = 0.0f;                 a.y = 0.0f; }
    // B (4x16, KxN): VGPR0 lanes0-15 = row K=0 (ones), lanes16-31 = row K=1 (area_n)
    v2f bb;
    bb.x = (lane < 16) ? 1.0f : area[tn * 16 + (lane - 16)];
    bb.y = 0.0f;
    v8f c = {};
    // D[m][n] = area_m + area_n   (matrix pipe does the pairwise broadcast-add)
    v8f d = __builtin_amdgcn_wmma_f32_16x16x4_f32(false, a, false, bb,
                                                  (short)0, c, false, false);

    // stage the 16 row boxes + 16 col boxes in LDS
    __shared__ float rb[16][4];
    __shared__ float cb[16][4];
    if (lane < 16) {
        const float* s = candBoxes + ((size_t)b * NCAND + tm * 16 + lane) * 4;
        rb[lane][0] = s[0]; rb[lane][1] = s[1]; rb[lane][2] = s[2]; rb[lane][3] = s[3];
    } else {
        const float* s = candBoxes + ((size_t)b * NCAND + tn * 16 + (lane - 16)) * 4;
        cb[lane - 16][0] = s[0]; cb[lane - 16][1] = s[1];
        cb[lane - 16][2] = s[2]; cb[lane - 16][3] = s[3];
    }
    __syncthreads();

    const int nloc = lane & 15;
    const float nx1 = cb[nloc][0], ny1 = cb[nloc][1], nx2 = cb[nloc][2], ny2 = cb[nloc][3];

    u32 bal[8];
#pragma unroll
    for (int v = 0; v < 8; ++v) {
        int mloc = (lane < 16) ? v : v + 8;      // C/D layout: VGPR v -> M=v / M=v+8
        float mx1 = rb[mloc][0], my1 = rb[mloc][1], mx2 = rb[mloc][2], my2 = rb[mloc][3];
        float w = fminf(mx2, nx2) - fmaxf(mx1, nx1);
        float h = fminf(my2, ny2) - fmaxf(my1, ny1);
        w = fmaxf(w, 0.0f);
        h = fmaxf(h, 0.0f);
        float inter = w * h;
        float uni   = d[v] - inter;              // area_m + area_n from the WMMA tile
        float iou   = inter / fmaxf(uni, 1e-9f);
        int   gm    = tm * 16 + mloc;
        int   gn    = tn * 16 + nloc;
        bool pred   = (iou > IOU_TH) && (gm != gn);   // diag zeroed in reference
        bal[v] = (u32)__ballot(pred);            // wave32: low 32 bits, uniform
    }
    // rows 0..7 are bal[v]&0xFFFF, rows 8..15 are bal[v]>>16 ; one u16 store per row
    if (lane < 16) {
        u32 bits = (lane < 8) ? (bal[lane] & 0xFFFFu) : ((bal[lane - 8] >> 16) & 0xFFFFu);
        int m = tm * 16 + lane;
        bitmat16[((size_t)b * NCAND + m) * 64 + tn] = (u16)bits;
    }
}

// ---------------- kernel 4: greedy NMS scan + top-750 emit, one wave per batch ---
__global__ __launch_bounds__(32) void k_nms_emit(const float* __restrict__ candBoxes,
                                                 const float* __restrict__ candScores,
                                                 const u32* __restrict__ bitmat32, // [B,1024,32]
                                                 float* __restrict__ outBoxes,    // [B,750,4]
                                                 float* __restrict__ outScores,   // [B,750]
                                                 float* __restrict__ outLabels,   // [B,750]
                                                 float* __restrict__ outMask) {   // [B,750]
    const int b    = blockIdx.x;
    const int lane = threadIdx.x;

    // zero-fill this batch's output slice (d_out is poisoned by the harness)
    for (int t = lane; t < NTOPK; t += 32) {
        float* ob = outBoxes + ((size_t)b * NTOPK + t) * 4;
        ob[0] = 0.0f; ob[1] = 0.0f; ob[2] = 0.0f; ob[3] = 0.0f;
        outScores[b * NTOPK + t] = 0.0f;
        outLabels[b * NTOPK + t] = 0.0f;
        outMask  [b * NTOPK + t] = 0.0f;
    }
    asm volatile("s_wait_storecnt 0" ::: "memory");  // order zero-fill vs. compacted writes

    // lane L owns candidates [L*32, L*32+32) as bit-words
    u32 validbits = 0;
#pragma unroll 4
    for (int k = 0; k < 32; ++k) {
        float s = candScores[b * NCAND + lane * 32 + k];
        if (s > CONF_TH) validbits |= (1u << k);
    }
    u32 supp = 0, keepbits = 0;

    for (int i = 0; i < NCAND; ++i) {
        int owner = i >> 5, bit = i & 31;
        u32 sw = __shfl(supp,      owner, 32);
        u32 vw = __shfl(validbits, owner, 32);
        bool keep_i = !((sw >> bit) & 1u) && ((vw >> bit) & 1u);
        if (keep_i) {
            supp |= bitmat32[((size_t)b * NCAND + i) * 32 + lane];
            if (lane == owner) keepbits |= (1u << bit);
        }
    }

    // wave-exclusive prefix sum of kept counts
    u32 cnt = (u32)__popc(keepbits);
    u32 pre = 0;
    for (int l = 0; l < 32; ++l) {
        u32 c = __shfl(cnt, l, 32);
        if (l < lane) pre += c;
    }
    // compact kept candidates (already score-descending) into first <=750 slots
    for (int k = 0; k < 32; ++k) {
        if ((keepbits >> k) & 1u) {
            u32 rank = (u32)__popc(keepbits & ((1u << k) - 1u));
            u32 pos  = pre + rank;
            if (pos < NTOPK) {
                int c = lane * 32 + k;
                const float* cbx = candBoxes + ((size_t)b * NCAND + c) * 4;
                float* ob = outBoxes + ((size_t)b * NTOPK + pos) * 4;
                ob[0] = cbx[0]; ob[1] = cbx[1]; ob[2] = cbx[2]; ob[3] = cbx[3];
                outScores[b * NTOPK + pos] = candScores[b * NCAND + c];
                outLabels[b * NTOPK + pos] = 1.0f;   // single foreground class
                outMask  [b * NTOPK + pos] = 1.0f;
            }
        }
    }
}

// ---------------- host-side launch ----------------
extern "C" void kernel_launch(void* const* d_in, const int* in_sizes, int n_in,
                              void* d_out, int out_size, void* d_ws, size_t ws_size,
                              hipStream_t stream) {
    const float* boxes_logits = (const float*)d_in[0];  // [16,21824,4]
    const float* cls_logits   = (const float*)d_in[1];  // [16,21824,2]
    const float* priors       = (const float*)d_in[2];  // [21824,4]

    // workspace layout (floats / u32, all 4B-aligned)
    float* boxesDec   = (float*)d_ws;                                   // B*P*4
    float* scoresM    = boxesDec   + (size_t)BATCH * NPRIOR * 4;        // B*P
    float* candBoxes  = scoresM    + (size_t)BATCH * NPRIOR;            // B*1024*4
    float* candScores = candBoxes  + (size_t)BATCH * NCAND * 4;         // B*1024
    float* candArea   = candScores + (size_t)BATCH * NCAND;             // B*1024
    u32*   bitmat32   = (u32*)(candArea + (size_t)BATCH * NCAND);       // B*1024*32
    u16*   bitmat16   = (u16*)bitmat32;

    float* outBoxes  = (float*)d_out;                         // [16,750,4]
    float* outScores = outBoxes  + (size_t)BATCH * NTOPK * 4; // [16,750]
    float* outLabels = outScores + (size_t)BATCH * NTOPK;     // [16,750]
    float* outMask   = outLabels + (size_t)BATCH * NTOPK;     // [16,750]

    int total = BATCH * NPRIOR;
    k_decode_score<<<(total + 255) / 256, 256, 0, stream>>>(
        boxes_logits, cls_logits, priors, boxesDec, scoresM);

    k_select<<<BATCH, 256, 0, stream>>>(scoresM, boxesDec,
                                        candBoxes, candScores, candArea);

    dim3 tgrid(NCAND / 16, NCAND / 16, BATCH);   // 64 x 64 x 16 tiles, 1 wave each
    k_iou_tiles<<<tgrid, 32, 0, stream>>>(candBoxes, candArea, bitmat16);

    k_nms_emit<<<BATCH, 32, 0, stream>>>(candBoxes, candScores, bitmat32,
                                         outBoxes, outScores, outLabels, outMask);
}